// MSF_DTA_49804440764953
// MI455X (gfx1250) — compile-verified
//
#include <hip/hip_runtime.h>

// Problem constants (match reference)
#define NNODES 98304
#define NEDGES 393216
#define NGRAPH 4096

typedef float v2f __attribute__((ext_vector_type(2)));
typedef float v8f __attribute__((ext_vector_type(8)));

// ---------------------------------------------------------------------------
// Degree / normalization:  deg[i] = in_degree + 1 (self loop); dinv = rsqrt
// ---------------------------------------------------------------------------
__global__ void deg_init(float* __restrict__ deg) {
    int i = blockIdx.x * blockDim.x + threadIdx.x;
    if (i < NNODES) deg[i] = 1.0f;   // self-loop contribution
}

__global__ void deg_count(const int* __restrict__ dst, float* __restrict__ deg) {
    int e = blockIdx.x * blockDim.x + threadIdx.x;
    if (e < NEDGES) unsafeAtomicAdd(&deg[dst[e]], 1.0f);  // global_atomic_add_f32
}

__global__ void deg_finalize(float* __restrict__ dinv) {
    int i = blockIdx.x * blockDim.x + threadIdx.x;
    if (i < NNODES) dinv[i] = rsqrtf(dinv[i]);            // deg >= 1 always
}

// ---------------------------------------------------------------------------
// GCN aggregation. Buffers hold PRE-relu values; consumers apply relu.
//   agg_init : out[i,f] = bias[f] + dinv[i]^2 * t[i,f]        (self loop)
//   agg_edges: out[d,f] += dinv[s]*dinv[d] * t[s,f]           (per edge)
// All indices fit i32 (max 30.7M elements); F/CHUNKS templated.
// ---------------------------------------------------------------------------
template <int F>
__global__ void agg_init(float* __restrict__ out, const float* __restrict__ t,
                         const float* __restrict__ bias, const float* __restrict__ dinv) {
    int idx = blockIdx.x * blockDim.x + threadIdx.x;
    if (idx >= NNODES * F) return;
    int i = idx / F;
    int f = idx % F;
    float di = dinv[i];
    out[idx] = bias[f] + di * di * t[idx];
}

// One wave handles one (edge, 32-feature chunk): coalesced gather + atomic scatter.
template <int F, int CHUNKS>
__global__ void agg_edges(float* __restrict__ out, const float* __restrict__ t,
                          const int* __restrict__ src, const int* __restrict__ dst,
                          const float* __restrict__ dinv) {
    int gid  = blockIdx.x * blockDim.x + threadIdx.x;   // < 126M, fits i32
    int wid  = gid >> 5;
    int lane = gid & 31;
    int e    = wid / CHUNKS;
    int ch   = wid % CHUNKS;
    if (e >= NEDGES) return;
    int f = ch * 32 + lane;
    if (f >= F) return;
    int s = src[e], d = dst[e];
    float coef = dinv[s] * dinv[d];
    unsafeAtomicAdd(&out[d * F + f], t[s * F + f] * coef);
}

// ---------------------------------------------------------------------------
// Global max-pool per graph. batch[] is sorted; binary search segment bounds.
// Buffer holds pre-relu values; init m=0 gives relu(max) = max(relu) for free.
// ---------------------------------------------------------------------------
__global__ void pool_max(const float* __restrict__ h, const int* __restrict__ batch,
                         float* __restrict__ g) {
    int b = blockIdx.x;
    int lo = 0, hi = NNODES;
    while (lo < hi) { int mid = (lo + hi) >> 1; if (batch[mid] < b) lo = mid + 1; else hi = mid; }
    int start = lo;
    hi = NNODES;
    while (lo < hi) { int mid = (lo + hi) >> 1; if (batch[mid] < b + 1) lo = mid + 1; else hi = mid; }
    int end = lo;
    for (int f = threadIdx.x; f < 312; f += blockDim.x) {
        float m = 0.0f;
        for (int i = start; i < end; ++i) m = fmaxf(m, h[i * 312 + f]);
        g[b * 312 + f] = m;
    }
}

// ---------------------------------------------------------------------------
// fp32 WMMA GEMM:  C[M,NC] = act( (reluIn?relu(A):A)[M,K] @ W[K,NC] + bias )
// K, NC, LDC are compile-time: unrolled-step byte offsets (k0*NC*4 <= 4MB)
// fold into the 24-bit instruction IOFFSET, and all per-lane offsets are i32
// (GVS addressing: saddr64 + voffset32 + imm), so the K-loop carries no
// address arithmetic at all.  f32 WMMA does NOT co-execute with VALU (only
// <=16-bit WMMA is TRANS-tracked), so stripping VALU from the loop directly
// feeds the matrix pipe.
// Assumptions (hold for every call site): M % 64 == 0; K even.
// Each wave owns a 32x32 C tile (2x2 v8f accumulators); block = 4 waves
// (2x2) covering a 64x64 C tile.
// ---------------------------------------------------------------------------
template <bool RELU_IN, bool BIAS, bool RELU_OUT, int K, int NC, int LDC>
__global__ __launch_bounds__(128)
void gemm_wmma(const float* __restrict__ A, const float* __restrict__ W,
               const float* __restrict__ bias, float* __restrict__ C, int M) {
    const int lane = threadIdx.x & 31;
    const int wave = threadIdx.x >> 5;
    const int row0 = blockIdx.y * 64 + (wave >> 1) * 32;
    const int col0 = blockIdx.x * 64 + (wave & 1) * 32;

    const int m  = lane & 15;
    const int kh = (lane >> 4) * 2;          // 0 or 2 (K split across half-waves)

    const int c0  = col0 + m;
    const int c1  = col0 + 16 + m;
    const int cl0 = c0 < NC ? c0 : NC - 1;   // clamp: loads unconditional
    const int cl1 = c1 < NC ? c1 : NC - 1;

    const int aoff0 = (row0 + m) * K + kh;   // i32, even -> float2 loads 8B aligned
    const int aoff1 = aoff0 + 16 * K;
    const int boff0 = kh * NC + cl0;
    const int boff1 = kh * NC + cl1;

    const v8f vzero = {0.f, 0.f, 0.f, 0.f, 0.f, 0.f, 0.f, 0.f};
    v8f acc00 = vzero, acc01 = vzero, acc10 = vzero, acc11 = vzero;

    constexpr int Kmain = K & ~3;
#pragma unroll 8
    for (int k0 = 0; k0 < Kmain; k0 += 4) {
        float2 fa0 = *(const float2*)(A + aoff0 + k0);   // global_load_b64
        float2 fa1 = *(const float2*)(A + aoff1 + k0);
        float b00 = W[boff0 + k0 * NC];                  // imm-offset, coalesced
        float b10 = W[boff1 + k0 * NC];
        float b01 = W[boff0 + k0 * NC + NC];
        float b11 = W[boff1 + k0 * NC + NC];
        v2f a0, a1, b0, b1;
        if (RELU_IN) {
            a0.x = fmaxf(fa0.x, 0.f); a0.y = fmaxf(fa0.y, 0.f);
            a1.x = fmaxf(fa1.x, 0.f); a1.y = fmaxf(fa1.y, 0.f);
        } else {
            a0.x = fa0.x; a0.y = fa0.y;
            a1.x = fa1.x; a1.y = fa1.y;
        }
        b0.x = b00; b0.y = b01;
        b1.x = b10; b1.y = b11;
        acc00 = __builtin_amdgcn_wmma_f32_16x16x4_f32(false, a0, false, b0, (short)0, acc00, false, false);
        acc01 = __builtin_amdgcn_wmma_f32_16x16x4_f32(false, a0, false, b1, (short)0, acc01, false, false);
        acc10 = __builtin_amdgcn_wmma_f32_16x16x4_f32(false, a1, false, b0, (short)0, acc10, false, false);
        acc11 = __builtin_amdgcn_wmma_f32_16x16x4_f32(false, a1, false, b1, (short)0, acc11, false, false);
    }

    if constexpr ((K & 3) != 0) {            // K tail (only K=78: remainder 2)
        const int ka = Kmain + kh;
        const bool k0ok = ka < K, k1ok = ka + 1 < K;
        v2f a0, a1, b0, b1;
        a0.x = k0ok ? A[aoff0 + Kmain] : 0.f;     a0.y = k1ok ? A[aoff0 + Kmain + 1] : 0.f;
        a1.x = k0ok ? A[aoff1 + Kmain] : 0.f;     a1.y = k1ok ? A[aoff1 + Kmain + 1] : 0.f;
        if (RELU_IN) {
            a0.x = fmaxf(a0.x, 0.f); a0.y = fmaxf(a0.y, 0.f);
            a1.x = fmaxf(a1.x, 0.f); a1.y = fmaxf(a1.y, 0.f);
        }
        b0.x = k0ok ? W[boff0 + Kmain * NC] : 0.f;  b0.y = k1ok ? W[boff0 + Kmain * NC + NC] : 0.f;
        b1.x = k0ok ? W[boff1 + Kmain * NC] : 0.f;  b1.y = k1ok ? W[boff1 + Kmain * NC + NC] : 0.f;
        acc00 = __builtin_amdgcn_wmma_f32_16x16x4_f32(false, a0, false, b0, (short)0, acc00, false, false);
        acc01 = __builtin_amdgcn_wmma_f32_16x16x4_f32(false, a0, false, b1, (short)0, acc01, false, false);
        acc10 = __builtin_amdgcn_wmma_f32_16x16x4_f32(false, a1, false, b0, (short)0, acc10, false, false);
        acc11 = __builtin_amdgcn_wmma_f32_16x16x4_f32(false, a1, false, b1, (short)0, acc11, false, false);
    }

    const bool ok0 = c0 < NC, ok1 = c1 < NC;
    const float bb0 = (BIAS && ok0) ? bias[c0] : 0.f;
    const float bb1 = (BIAS && ok1) ? bias[c1] : 0.f;
    const int rb = row0 + ((lane >> 4) * 8);
#pragma unroll
    for (int r = 0; r < 8; ++r) {
        if (ok0) {
            float v = acc00[r] + bb0; if (RELU_OUT) v = fmaxf(v, 0.f);
            C[(rb + r) * LDC + c0] = v;
            v = acc10[r] + bb0;       if (RELU_OUT) v = fmaxf(v, 0.f);
            C[(rb + 16 + r) * LDC + c0] = v;
        }
        if (ok1) {
            float v = acc01[r] + bb1; if (RELU_OUT) v = fmaxf(v, 0.f);
            C[(rb + r) * LDC + c1] = v;
            v = acc11[r] + bb1;       if (RELU_OUT) v = fmaxf(v, 0.f);
            C[(rb + 16 + r) * LDC + c1] = v;
        }
    }
}

#define GEMM_GRID(M, Nc) dim3((unsigned)(((Nc) + 63) / 64), (unsigned)((M) / 64))

extern "C" void kernel_launch(void* const* d_in, const int* in_sizes, int n_in,
                              void* d_out, int out_size, void* d_ws, size_t ws_size,
                              hipStream_t stream) {
    const float* x      = (const float*)d_in[0];
    const int*   ei     = (const int*)  d_in[1];
    const int*   batch  = (const int*)  d_in[2];
    const float* target = (const float*)d_in[3];
    // d_in[4] fingerprint, d_in[21] Wfp, d_in[22] bfp: dead in the reference (fp deleted)
    const float* W1  = (const float*)d_in[5];   const float* b1  = (const float*)d_in[6];
    const float* W2  = (const float*)d_in[7];   const float* b2  = (const float*)d_in[8];
    const float* W3  = (const float*)d_in[9];   const float* b3  = (const float*)d_in[10];
    const float* Wg1 = (const float*)d_in[11];  const float* bg1 = (const float*)d_in[12];
    const float* Wg2 = (const float*)d_in[13];  const float* bg2 = (const float*)d_in[14];
    const float* Wd  = (const float*)d_in[15];  const float* bd  = (const float*)d_in[16];
    const float* Wd2 = (const float*)d_in[17];  const float* bd2 = (const float*)d_in[18];
    const float* Wd3 = (const float*)d_in[19];  const float* bd3 = (const float*)d_in[20];
    const float* Wf1 = (const float*)d_in[23];  const float* bf1 = (const float*)d_in[24];
    const float* Wf2 = (const float*)d_in[25];  const float* bf2 = (const float*)d_in[26];
    const float* Wo  = (const float*)d_in[27];  const float* bo  = (const float*)d_in[28];

    const int* src = ei;             // edge_index[0]
    const int* dst = ei + NEDGES;    // edge_index[1]

    // Workspace layout (~246 MB): dinv | bufT (N*312) | bufU (N*312).
    // B-side buffers reuse bufT once the GCN transforms are finished.
    float* dinv = (float*)d_ws;
    float* bufT = dinv + NNODES;
    float* bufU = bufT + (long)NNODES * 312;
    float* g0  = bufT;
    float* g1  = g0  + (long)NGRAPH * 312;
    float* xt1 = g1  + (long)NGRAPH * 1024;
    float* xt2 = xt1 + (long)NGRAPH * 512;
    float* xc  = xt2 + (long)NGRAPH * 256;   // [B,256] = concat(g, xt)
    float* f1  = xc  + (long)NGRAPH * 256;
    float* f2  = f1  + (long)NGRAPH * 1024;

    // --- sym-normalization coefficients ---
    deg_init    <<<(NNODES + 255) / 256, 256, 0, stream>>>(dinv);
    deg_count   <<<(NEDGES + 255) / 256, 256, 0, stream>>>(dst, dinv);
    deg_finalize<<<(NNODES + 255) / 256, 256, 0, stream>>>(dinv);

    #define AGG_INIT_BLOCKS(F) ((unsigned)(((long)NNODES * (F) + 255) / 256))
    #define AGG_EDGE_BLOCKS(CH) ((unsigned)(((long)NEDGES * (CH) * 32 + 255) / 256))

    // --- GCN layer 1: x@W1 -> aggregate (+b1); relu deferred to consumer ---
    gemm_wmma<false, false, false, 78, 78, 78><<<GEMM_GRID(NNODES, 78), 128, 0, stream>>>(
        x, W1, nullptr, bufT, NNODES);
    agg_init<78>     <<<AGG_INIT_BLOCKS(78), 256, 0, stream>>>(bufU, bufT, b1, dinv);
    agg_edges<78, 3> <<<AGG_EDGE_BLOCKS(3), 256, 0, stream>>>(bufU, bufT, src, dst, dinv);

    // --- GCN layer 2 ---
    gemm_wmma<true, false, false, 78, 156, 156><<<GEMM_GRID(NNODES, 156), 128, 0, stream>>>(
        bufU, W2, nullptr, bufT, NNODES);
    agg_init<156>    <<<AGG_INIT_BLOCKS(156), 256, 0, stream>>>(bufU, bufT, b2, dinv);
    agg_edges<156, 5><<<AGG_EDGE_BLOCKS(5), 256, 0, stream>>>(bufU, bufT, src, dst, dinv);

    // --- GCN layer 3 ---
    gemm_wmma<true, false, false, 156, 312, 312><<<GEMM_GRID(NNODES, 312), 128, 0, stream>>>(
        bufU, W3, nullptr, bufT, NNODES);
    agg_init<312>     <<<AGG_INIT_BLOCKS(312), 256, 0, stream>>>(bufU, bufT, b3, dinv);
    agg_edges<312, 10><<<AGG_EDGE_BLOCKS(10), 256, 0, stream>>>(bufU, bufT, src, dst, dinv);

    // --- global max pool (applies relu implicitly via m=0 init) ---
    pool_max<<<NGRAPH, 128, 0, stream>>>(bufU, batch, g0);

    // --- graph MLP: g = relu(g0@Wg1+bg1)@Wg2+bg2 -> xc[:, 0:128] ---
    gemm_wmma<false, true, true, 312, 1024, 1024><<<GEMM_GRID(NGRAPH, 1024), 128, 0, stream>>>(
        g0, Wg1, bg1, g1, NGRAPH);
    gemm_wmma<false, true, false, 1024, 128, 256><<<GEMM_GRID(NGRAPH, 128), 128, 0, stream>>>(
        g1, Wg2, bg2, xc, NGRAPH);

    // --- target MLP -> xc[:, 128:256] ---
    gemm_wmma<false, true, true, 800, 512, 512><<<GEMM_GRID(NGRAPH, 512), 128, 0, stream>>>(
        target, Wd, bd, xt1, NGRAPH);
    gemm_wmma<false, true, true, 512, 256, 256><<<GEMM_GRID(NGRAPH, 256), 128, 0, stream>>>(
        xt1, Wd2, bd2, xt2, NGRAPH);
    gemm_wmma<false, true, false, 256, 128, 256><<<GEMM_GRID(NGRAPH, 128), 128, 0, stream>>>(
        xt2, Wd3, bd3, xc + 128, NGRAPH);

    // --- head ---
    gemm_wmma<false, true, true, 256, 1024, 1024><<<GEMM_GRID(NGRAPH, 1024), 128, 0, stream>>>(
        xc, Wf1, bf1, f1, NGRAPH);
    gemm_wmma<false, true, true, 1024, 512, 512><<<GEMM_GRID(NGRAPH, 512), 128, 0, stream>>>(
        f1, Wf2, bf2, f2, NGRAPH);
    gemm_wmma<false, true, false, 512, 1, 1><<<GEMM_GRID(NGRAPH, 1), 128, 0, stream>>>(
        f2, Wo, bo, (float*)d_out, NGRAPH);
}